// RoboNodeEncoder_88424786690347
// MI455X (gfx1250) — compile-verified
//
#include <hip/hip_runtime.h>
#include <math.h>

typedef __attribute__((ext_vector_type(2))) float v2f;
typedef __attribute__((ext_vector_type(8))) float v8f;

#define EPSV 1e-5f

// ---------------------------------------------------------------------------
// float atomic max via sign-split ordered-int trick (lowers to
// global_atomic_max_i32 / global_atomic_min_u32 — no CAS loop).
// Works with buffer initialized to -inf.
// ---------------------------------------------------------------------------
__device__ __forceinline__ void atomicMaxFloat(float* addr, float val) {
    if (val >= 0.0f) {
        atomicMax((int*)addr, __float_as_int(val));
    } else {
        atomicMin((unsigned int*)addr, __float_as_uint(val));
    }
}

// ---------------------------------------------------------------------------
// WMMA f32 16x16x4 (CDNA5 ISA §7.12.2 layouts):
//   A 16x4 : lanes 0-15 hold M=lane, K={0,1}; lanes 16-31 hold M=lane-16, K={2,3}
//   B 4x16 : lanes 0-15 hold N=lane, K={0,1}; lanes 16-31 hold N=lane-16, K={2,3}
//   C/D    : VGPR r -> M = r + 8*(lane>=16), N = lane&15
// ---------------------------------------------------------------------------
__device__ __forceinline__ v8f wmma_f32(v2f a, v2f b, v8f c) {
    return __builtin_amdgcn_wmma_f32_16x16x4_f32(false, a, false, b, (short)0, c,
                                                 false, false);
}

// Async global->LDS copy of 16B, per-lane LDS dest (ASYNCcnt path, ISA §10/§08).
__device__ __forceinline__ void async_copy_b128(unsigned lds_byte_off,
                                                const void* gptr) {
    unsigned long long ga = (unsigned long long)gptr;
    asm volatile("global_load_async_to_lds_b128 %0, %1, off"
                 :: "v"(lds_byte_off), "v"(ga)
                 : "memory");
}
__device__ __forceinline__ void wait_asynccnt0() {
    asm volatile("s_wait_asynccnt 0x0" ::: "memory");
}
__device__ __forceinline__ unsigned lds_addr_of(const void* p) {
    return (unsigned)(uintptr_t)p;  // LDS flat addr: offset lives in addr[31:0]
}

// ---------------------------------------------------------------------------
__global__ __launch_bounds__(256) void k_fill_neginf(float* p, int n) {
    int i = blockIdx.x * 256 + threadIdx.x;
    if (i < n) p[i] = -INFINITY;
}

// -inf (empty segment) -> 0, matching jnp.where(isfinite(xg), xg, 0)
__global__ __launch_bounds__(256) void k_sanitize(float* p, int n) {
    int i = blockIdx.x * 256 + threadIdx.x;
    if (i < n) {
        float v = p[i];
        p[i] = isfinite(v) ? v : 0.0f;
    }
}

// ---------------------------------------------------------------------------
// Fused PointNet mlp1 + scatter-max.
//   per point: h = relu(LN(x @ w1 + b1)) ; y = h @ w2 + b2 ; xg[idx] = max(...)
// 8 waves / block, 16 points per wave. w2 staged pair-interleaved so every
// B fragment is one aligned ds_load_b64.
// ---------------------------------------------------------------------------
__global__ __launch_bounds__(256) void k_mlp1_scatter(
    const float* __restrict__ X,   // [P,3]
    const int* __restrict__ Idx,   // [P]
    const float* __restrict__ w1,  // [3,64]
    const float* __restrict__ b1,  // [64]
    const float* __restrict__ g1,  // [64]
    const float* __restrict__ be1, // [64]
    const float* __restrict__ w2,  // [64,64]
    const float* __restrict__ b2,  // [64]
    float* __restrict__ xg)        // [NN,64], pre-filled with -inf
{
    __shared__ float w1s[3 * 64];
    __shared__ float b1s[64], g1s[64], be1s[64], b2s[64];
    // pair-interleaved w2: element (k,n) at [(k>>1)*160 + n*2 + (k&1)]
    // (pair-row stride 160 = 128+32 keeps lane-half bank sets disjoint)
    __shared__ float w2i[32 * 160];
    __shared__ float hs[8][16 * 68];  // per-wave 16x64 tile, stride 68

    const int tid  = threadIdx.x;
    const int lane = tid & 31;
    const int w    = tid >> 5;       // wave id 0..7
    const int p    = lane & 15;      // row within 16-tile
    const int half = lane >> 4;      // 0/1

    for (int i = tid; i < 64 * 64; i += 256) {
        const int k = i >> 6, n = i & 63;
        w2i[(k >> 1) * 160 + n * 2 + (k & 1)] = w2[i];
    }
    for (int i = tid; i < 3 * 64; i += 256) w1s[i] = w1[i];
    for (int i = tid; i < 64; i += 256) {
        b1s[i] = b1[i]; g1s[i] = g1[i]; be1s[i] = be1[i]; b2s[i] = b2[i];
    }
    __syncthreads();

    const long base = (long)blockIdx.x * 128 + (long)w * 16;  // first point of tile
    const long g    = base + p;
    const int  myidx = Idx[base + p];   // node id of point p (lanes 16-31 duplicate)

    // ---- mlp1 layer 1 + LN + relu (lane pair per point) -------------------
    const float x0 = X[g * 3 + 0];
    const float x1 = X[g * 3 + 1];
    const float x2 = X[g * 3 + 2];

    float h[32];
    float s1 = 0.f, s2 = 0.f;
#pragma unroll
    for (int j = 0; j < 32; ++j) {
        int f   = half * 32 + j;
        float v = fmaf(x0, w1s[f], fmaf(x1, w1s[64 + f], fmaf(x2, w1s[128 + f], b1s[f])));
        h[j] = v;
        s1 += v;
        s2 += v * v;
    }
    s1 += __shfl_xor(s1, 16, 32);
    s2 += __shfl_xor(s2, 16, 32);
    const float mean = s1 * (1.0f / 64.0f);
    const float var  = s2 * (1.0f / 64.0f) - mean * mean;
    const float rs   = rsqrtf(var + EPSV);

    float* hrow = &hs[w][p * 68];
#pragma unroll
    for (int j = 0; j < 32; ++j) {
        int f   = half * 32 + j;
        float v = (h[j] - mean) * rs * g1s[f] + be1s[f];
        hrow[f] = fmaxf(v, 0.0f);
    }
    __syncthreads();

    // ---- 16x64x64 GEMM via WMMA f32 ---------------------------------------
    v8f acc[4];
#pragma unroll
    for (int nt = 0; nt < 4; ++nt) {
        float bb = b2s[nt * 16 + p];
#pragma unroll
        for (int r = 0; r < 8; ++r) acc[nt][r] = bb;
    }

    const float* ht = &hs[w][0];
#pragma unroll 4
    for (int k0 = 0; k0 < 64; k0 += 4) {
        const int kb = k0 + 2 * half;  // even
        const v2f a = *(const v2f*)&ht[p * 68 + kb];
#pragma unroll
        for (int nt = 0; nt < 4; ++nt) {
            const v2f b = *(const v2f*)&w2i[(kb >> 1) * 160 + (nt * 16 + p) * 2];
            acc[nt] = wmma_f32(a, b, acc[nt]);
        }
    }

    // ---- atomic scatter-max -----------------------------------------------
#pragma unroll
    for (int r = 0; r < 8; ++r) {
        const int m    = r + 8 * half;
        const int node = __shfl(myidx, m, 32);
        float* xr      = xg + (long)node * 64;
#pragma unroll
        for (int nt = 0; nt < 4; ++nt) atomicMaxFloat(&xr[nt * 16 + p], acc[nt][r]);
    }
}

// ---------------------------------------------------------------------------
// Generic fp32 WMMA GEMM:  C[:, coloff:coloff+N] = A[M,K] @ B[K,N] + bias
// Block tile 64x128, 8 waves (4x2), each wave a 16x64 sub-tile (4 accs).
// A tile staged via global_load_async_to_lds_b128 (per-lane LDS scatter keeps
// the 80-byte padded rows 16B-aligned); B staged pair-interleaved so each
// fragment is one ds_load_b64. Requires M%64==0, N%128==0, K%16==0 (true for
// every GEMM here), so EXEC is all-ones for every v_wmma.
// ---------------------------------------------------------------------------
__global__ __launch_bounds__(256) void k_gemm_wmma(
    const float* __restrict__ A, const float* __restrict__ B,
    const float* __restrict__ bias, float* __restrict__ C,
    int K, int N, int ldc, int coloff)
{
    __shared__ float As[64 * 20];   // 64 rows x 16, stride 20 (80B rows)
    __shared__ float Bs[8 * 288];   // pair-interleaved, pair-row stride 288

    const int tid  = threadIdx.x;
    const int lane = tid & 31;
    const int w    = tid >> 5;
    const int m0b  = blockIdx.y * 64;
    const int n0b  = blockIdx.x * 128;
    const int m0w  = (w >> 1) * 16;
    const int n0w  = (w & 1) * 64;
    const int p    = lane & 15;
    const int half = lane >> 4;

    v8f acc[4];
#pragma unroll
    for (int nt = 0; nt < 4; ++nt) {
        float bb = bias[n0b + n0w + nt * 16 + p];
#pragma unroll
        for (int r = 0; r < 8; ++r) acc[nt][r] = bb;
    }

    const int arow = tid >> 2;  // 0..63
    const int aq   = tid & 3;   // 0..3 (float4 within 16-wide chunk)
    const unsigned a_lds = lds_addr_of(&As[arow * 20 + aq * 4]);
    const float* agp     = A + (long)(m0b + arow) * K + aq * 4;

    for (int k0 = 0; k0 < K; k0 += 16) {
        __syncthreads();
        // stage A chunk (64x16): async 16B copy per thread into padded rows
        async_copy_b128(a_lds, agp + k0);
        // stage B chunk (16x128): coalesced float4 loads, interleaved stores
#pragma unroll
        for (int i = 0; i < 2; ++i) {
            const int idx = tid + i * 256;
            const int r   = idx >> 5;   // k row 0..15
            const int c   = idx & 31;   // float4 col group
            const float4 bv = *(const float4*)(B + (long)(k0 + r) * N + n0b + c * 4);
            float* d = &Bs[(r >> 1) * 288 + (r & 1) + c * 8];
            d[0] = bv.x; d[2] = bv.y; d[4] = bv.z; d[6] = bv.w;
        }
        if (k0 + 16 < K) {
            __builtin_prefetch(B + (long)(k0 + 16) * N + n0b, 0, 1);
        }
        wait_asynccnt0();
        __syncthreads();

#pragma unroll
        for (int kk = 0; kk < 16; kk += 4) {
            const int kb = kk + 2 * half;  // even
            const v2f a = *(const v2f*)&As[(m0w + p) * 20 + kb];
#pragma unroll
            for (int nt = 0; nt < 4; ++nt) {
                const v2f b = *(const v2f*)&Bs[(kb >> 1) * 288 + (n0w + nt * 16 + p) * 2];
                acc[nt] = wmma_f32(a, b, acc[nt]);
            }
        }
    }

#pragma unroll
    for (int nt = 0; nt < 4; ++nt) {
        const int n = n0b + n0w + nt * 16 + p;
#pragma unroll
        for (int r = 0; r < 8; ++r) {
            const int m = m0b + m0w + r + 8 * half;
            C[(long)m * ldc + coloff + n] = acc[nt][r];
        }
    }
}

// ---------------------------------------------------------------------------
// Row LayerNorm(768) * g + be, then ReLU. One wave per row (wave32 butterfly).
// ---------------------------------------------------------------------------
__global__ __launch_bounds__(256) void k_rowln_relu(
    const float* __restrict__ X, const float* __restrict__ g,
    const float* __restrict__ be, float* __restrict__ O, int D)
{
    const int lane = threadIdx.x & 31;
    const int w    = threadIdx.x >> 5;
    const long row = (long)blockIdx.x * 8 + w;
    const float* xr = X + row * D;

    float s1 = 0.f, s2 = 0.f;
    for (int j = lane; j < D; j += 32) {
        float v = xr[j];
        s1 += v;
        s2 += v * v;
    }
#pragma unroll
    for (int m = 16; m >= 1; m >>= 1) {
        s1 += __shfl_xor(s1, m, 32);
        s2 += __shfl_xor(s2, m, 32);
    }
    const float inv  = 1.0f / (float)D;
    const float mean = s1 * inv;
    const float var  = s2 * inv - mean * mean;
    const float rs   = rsqrtf(var + EPSV);

    float* orow = O + row * D;
    for (int j = lane; j < D; j += 32) {
        float v = (xr[j] - mean) * rs * g[j] + be[j];
        orow[j] = fmaxf(v, 0.0f);
    }
}

// ---------------------------------------------------------------------------
// Host-side orchestration
// ---------------------------------------------------------------------------
extern "C" void kernel_launch(void* const* d_in, const int* in_sizes, int n_in,
                              void* d_out, int out_size, void* d_ws, size_t ws_size,
                              hipStream_t stream) {
    const float* x_pos   = (const float*)d_in[0];
    const int*   pos_idx = (const int*)d_in[1];
    const float* x_aff   = (const float*)d_in[2];
    const int*   aff_idx = (const int*)d_in[3];
    const float* x_sem   = (const float*)d_in[4];
    // d_in[5] = num_nodes scalar (device); derive shapes host-side instead:
    const int P  = in_sizes[0] / 3;    // 1048576
    const int NN = in_sizes[4] / 512;  // 16384
    const int D  = 768;

    const float* sem_w  = (const float*)d_in[30];
    const float* sem_b  = (const float*)d_in[31];
    const float* fus_w  = (const float*)d_in[32];
    const float* fus_b  = (const float*)d_in[33];
    const float* fus_g  = (const float*)d_in[34];
    const float* fus_be = (const float*)d_in[35];

    // workspace layout
    float* ws = (float*)d_ws;
    float* xg = ws;                        // NN*64   (reused per prefix)
    float* tA = xg + (size_t)NN * 64;      // NN*768
    float* tB = tA + (size_t)NN * D;       // NN*768
    float* cc = tB + (size_t)NN * D;       // NN*2304 (concat buffer)

    const dim3 gGemm(D / 128, NN / 64);    // (6, 256)

    for (int pfx = 0; pfx < 2; ++pfx) {
        const int b = 6 + pfx * 12;
        const float* w1  = (const float*)d_in[b + 0];
        const float* b1  = (const float*)d_in[b + 1];
        const float* g1  = (const float*)d_in[b + 2];
        const float* be1 = (const float*)d_in[b + 3];
        const float* w2  = (const float*)d_in[b + 4];
        const float* b2  = (const float*)d_in[b + 5];
        const float* w3  = (const float*)d_in[b + 6];
        const float* b3  = (const float*)d_in[b + 7];
        const float* g2  = (const float*)d_in[b + 8];
        const float* be2 = (const float*)d_in[b + 9];
        const float* w4  = (const float*)d_in[b + 10];
        const float* b4  = (const float*)d_in[b + 11];
        const float* X   = pfx ? x_aff : x_pos;
        const int*   I   = pfx ? aff_idx : pos_idx;

        k_fill_neginf<<<(NN * 64 + 255) / 256, 256, 0, stream>>>(xg, NN * 64);
        k_mlp1_scatter<<<P / 128, 256, 0, stream>>>(X, I, w1, b1, g1, be1, w2, b2, xg);
        k_sanitize<<<(NN * 64 + 255) / 256, 256, 0, stream>>>(xg, NN * 64);
        // t = relu(LN(xg @ w3 + b3))
        k_gemm_wmma<<<gGemm, 256, 0, stream>>>(xg, w3, b3, tA, 64, D, D, 0);
        k_rowln_relu<<<NN / 8, 256, 0, stream>>>(tA, g2, be2, tB, D);
        // concat[:, pfx*768 : ] = t @ w4 + b4
        k_gemm_wmma<<<gGemm, 256, 0, stream>>>(tB, w4, b4, cc, D, D, 3 * D, pfx * D);
    }

    // concat[:, 1536:] = x_sem @ sem_w + sem_b
    k_gemm_wmma<<<gGemm, 256, 0, stream>>>(x_sem, sem_w, sem_b, cc, 512, D, 3 * D, 2 * D);
    // fused = concat @ fus_w + fus_b
    k_gemm_wmma<<<gGemm, 256, 0, stream>>>(cc, fus_w, fus_b, tA, 3 * D, D, D, 0);
    // out = relu(LN(fused))
    k_rowln_relu<<<NN / 8, 256, 0, stream>>>(tA, fus_g, fus_be, (float*)d_out, D);
}